// SoftContrastiveLoss_51092930953476
// MI455X (gfx1250) — compile-verified
//
#include <hip/hip_runtime.h>
#include <hip/hip_bf16.h>
#include <math.h>

// ---------------- problem constants (from reference) ----------------
#define NN      131072
#define DD      128
#define UU      1024
#define PP      256
#define TEMP    0.3f
#define LAMBDA_ORTHO 0.1f
#define LOWT    0.4f
#define HIGHT   0.7f
#define HINGE_M 0.1f
#define W_NEG   1.5f
#define W_UNL   0.3f
#define EPSN    1e-8f

#define CHUNK   64                    // rows of u/p staged per LDS pass (64KB total)
#define NCHUNKS (NN / CHUNK)          // 2048

#if __has_builtin(__builtin_amdgcn_global_load_async_to_lds_b128) && \
    __has_builtin(__builtin_amdgcn_s_wait_asynccnt)
#define USE_ASYNC_LDS 1
#else
#define USE_ASYNC_LDS 0
#endif

typedef __attribute__((ext_vector_type(2))) float v2f;
typedef __attribute__((ext_vector_type(8))) float v8f;

// exact parameter type of the async-LDS builtins on this toolchain:
//   '__attribute__((__vector_size__(4 * sizeof(int)))) int __device__ *'  (AS1, non-const)
typedef int vi4 __attribute__((vector_size(4 * sizeof(int))));
typedef __attribute__((address_space(1))) vi4* as1_vi4_p;
typedef __attribute__((address_space(3))) vi4* as3_vi4_p;

// ---------------- block reduction helper (wave32) ----------------
__device__ __forceinline__ float blockReduceSum256(float v, float* sred /* >=8 floats */) {
    #pragma unroll
    for (int o = 16; o > 0; o >>= 1) v += __shfl_xor(v, o);
    int lane = threadIdx.x & 31, wid = threadIdx.x >> 5;
    __syncthreads();               // protect sred across repeated calls
    if (lane == 0) sred[wid] = v;
    __syncthreads();
    float r = 0.f;
    if (threadIdx.x < 8) r = sred[threadIdx.x];
    if (wid == 0) {
        #pragma unroll
        for (int o = 4; o > 0; o >>= 1) r += __shfl_xor(r, o);
    }
    return r;  // thread 0 holds the total
}

// ---------------- K0: init workspace ----------------
__global__ void init_kernel(int* __restrict__ winner, float* __restrict__ scal) {
    int i = blockIdx.x * 256 + threadIdx.x;          // grid covers U*P = 262144
    winner[i] = -1;
    if (i < 4) scal[i] = 0.f;
}

// ---------------- K1: persistent fused dist + winner scatter + split-K fp32 WMMA GEMM ----
// Each block owns `cpb` contiguous 64-row chunks, accumulates its O-partial in registers,
// then writes a private 128x128 partial (NO atomics on O).
__global__ __launch_bounds__(256)
void fused_dist_gemm(const float* __restrict__ u, const float* __restrict__ p,
                     const int* __restrict__ uid, const int* __restrict__ pid,
                     float* __restrict__ dist, int* __restrict__ winner,
                     float* __restrict__ Opart, int cpb) {
    __shared__ float Ul[CHUNK * DD];   // 32 KB
    __shared__ float Pl[CHUNK * DD];   // 32 KB

    const int lane = threadIdx.x & 31;
    const int wv   = threadIdx.x >> 5;            // 8 waves -> 8 row tiles of output
    const int l16  = lane & 15;
    const int kr0  = (lane >> 4) * 2;             // A/B frag: k = (lane>>4)*2 + vgpr
    const int mcol = wv * 16 + l16;

    v8f acc[8];
    #pragma unroll
    for (int t = 0; t < 8; ++t) acc[t] = (v8f){0.f,0.f,0.f,0.f,0.f,0.f,0.f,0.f};

    float4* Ul4 = (float4*)Ul;
    float4* Pl4 = (float4*)Pl;

    for (int cc = 0; cc < cpb; ++cc) {
        const int chunkStart = (blockIdx.x * cpb + cc) * CHUNK;
        const size_t base = (size_t)chunkStart * DD;
        const float4* usrc = (const float4*)(u + base);
        const float4* psrc = (const float4*)(p + base);

        // ---- stage chunk into LDS ----
#if USE_ASYNC_LDS
        #pragma unroll
        for (int it = 0; it < (CHUNK * DD / 4) / 256; ++it) {
            int i = it * 256 + threadIdx.x;
            __builtin_amdgcn_global_load_async_to_lds_b128(
                (as1_vi4_p)(usrc + i), (as3_vi4_p)(Ul4 + i), 0, 0);
            __builtin_amdgcn_global_load_async_to_lds_b128(
                (as1_vi4_p)(psrc + i), (as3_vi4_p)(Pl4 + i), 0, 0);
        }
#else
        #pragma unroll
        for (int it = 0; it < (CHUNK * DD / 4) / 256; ++it) {
            int i = it * 256 + threadIdx.x;
            Ul4[i] = usrc[i];
            Pl4[i] = psrc[i];
        }
#endif
        // warm L2 with the next chunk while this one computes
        if (cc + 1 < cpb) {
            const float* un = u + base + (size_t)CHUNK * DD;
            const float* pn = p + base + (size_t)CHUNK * DD;
            __builtin_prefetch(un + threadIdx.x * 64, 0, 1);
            __builtin_prefetch(pn + threadIdx.x * 64, 0, 1);
        }
#if USE_ASYNC_LDS
        __builtin_amdgcn_s_wait_asynccnt(0);
#endif
        __syncthreads();

        // ---- pairwise distance for 64 rows (4 threads/row) + last-write scatter ----
        {
            int row  = threadIdx.x >> 2;
            int part = threadIdx.x & 3;
            float s = 0.f;
            int d0 = part * 32;
            #pragma unroll 8
            for (int dd = d0; dd < d0 + 32; ++dd) {
                float df = Ul[row * DD + dd] - Pl[row * DD + dd] + 1e-6f;
                s = fmaf(df, df, s);
            }
            s += __shfl_xor(s, 1);
            s += __shfl_xor(s, 2);
            if (part == 0) {
                int n = chunkStart + row;
                dist[n] = sqrtf(s);
                int cell = uid[n] * PP + pid[n];
                atomicMax(&winner[cell], n);      // last-write == max sample index
            }
        }

        // ---- O[d1][d2] += sum_k u[k][d1] * p[k][d2] via V_WMMA_F32_16X16X4_F32 ----
        for (int kb = 0; kb < CHUNK; kb += 4) {
            v2f a;
            a.x = Ul[(kb + kr0    ) * DD + mcol];
            a.y = Ul[(kb + kr0 + 1) * DD + mcol];
            #pragma unroll
            for (int t = 0; t < 8; ++t) {
                v2f b;
                int ncol = t * 16 + l16;
                b.x = Pl[(kb + kr0    ) * DD + ncol];
                b.y = Pl[(kb + kr0 + 1) * DD + ncol];
                acc[t] = __builtin_amdgcn_wmma_f32_16x16x4_f32(
                             false, a, false, b, (short)0, acc[t], false, false);
            }
        }
        __syncthreads();   // all waves done reading LDS before next stage
    }

    // ---- plain (non-atomic) partial store: one private 128x128 tile per block ----
    // C/D layout: m = vgpr + (lane>=16 ? 8 : 0), n = lane&15
    float* myPart = Opart + (size_t)blockIdx.x * (DD * DD);
    const int mofs = (lane >> 4) * 8;
    #pragma unroll
    for (int t = 0; t < 8; ++t) {
        #pragma unroll
        for (int v = 0; v < 8; ++v) {
            int row = wv * 16 + v + mofs;
            int col = t * 16 + l16;
            myPart[row * DD + col] = acc[t][v];
        }
    }
}

// ---------------- K2: fold partials -> sum |O| into scal[3] ----------------
__global__ __launch_bounds__(256)
void reduce_ortho(const float* __restrict__ Opart, float* __restrict__ scal, int split) {
    __shared__ float sred[8];
    int elem = blockIdx.x * 256 + threadIdx.x;       // grid covers 16384 elements
    float s = 0.f;
    for (int pp = 0; pp < split; ++pp)
        s += Opart[(size_t)pp * (DD * DD) + elem];   // coalesced across threads
    float tot = blockReduceSum256(fabsf(s), sred);
    if (threadIdx.x == 0) atomicAdd(&scal[3], tot);
}

// ---------------- K3: per-user InfoNCE + pairwise hinge (one block per user) ----------------
__global__ __launch_bounds__(256)
void loss_kernel(const float* __restrict__ t, const float* __restrict__ dist,
                 const int* __restrict__ winner, float* __restrict__ scal) {
    __shared__ float sM[PP];
    __shared__ float sD[PP];
    __shared__ float sred[8];

    const int urow = blockIdx.x;
    const int j0   = threadIdx.x;
    const int cell = urow * PP + j0;

    float Tv = 0.f, Mv = 0.f, Dv = 0.f;
    int w = winner[cell];
    if (w >= 0) {
        float tv = t[w];
        Mv = tv;
        Tv = (tv > HIGHT) ? 1.f : ((tv < LOWT) ? -1.f : 0.f);
        Dv = dist[w];
    }
    sM[j0] = Mv;
    sD[j0] = Dv;

    const bool  pos = (Tv == 1.f);
    const float Wt  = pos ? 1.f : ((Tv == -1.f) ? W_NEG : W_UNL);
    const float wgt = Wt * expf(-Dv / TEMP);

    float den  = blockReduceSum256(wgt, sred);
    float num  = blockReduceSum256(pos ? wgt : 0.f, sred);
    float npos = blockReduceSum256(pos ? 1.f : 0.f, sred);

    __syncthreads();   // sM/sD visible & sred quiesced

    // hinge over pairs i<j within this user's row
    float hsum = 0.f;
    if (Mv > 0.f) {
        for (int j = j0 + 1; j < PP; ++j) {
            float mj = sM[j];
            if (mj > 0.f && mj != Mv) {
                float sgn  = (mj > Mv) ? 1.f : -1.f;      // sign(M_j - M_i), nonzero
                float term = sgn * (Dv - sD[j]) + HINGE_M;
                hsum += fmaxf(term, 0.f);
            }
        }
    }
    float htot = blockReduceSum256(hsum, sred);

    if (threadIdx.x == 0) {
        bool valid = npos > 0.5f;
        float per_user = valid ? -logf(num / (den + EPSN)) : 0.f;
        atomicAdd(&scal[0], per_user);
        atomicAdd(&scal[1], valid ? 1.f : 0.f);
        atomicAdd(&scal[2], htot);
    }
}

// ---------------- K4: combine scalars ----------------
__global__ void final_kernel(const float* __restrict__ scal, float* __restrict__ out) {
    if (threadIdx.x == 0) {
        float nv    = scal[1];
        float info  = (nv > 0.5f) ? (scal[0] / nv) : 0.f;
        float hinge = scal[2] / (float)UU;
        float ortho = scal[3] / (float)(DD * DD);
        out[0] = info + hinge + ortho * LAMBDA_ORTHO;
    }
}

// ---------------- launch ----------------
extern "C" void kernel_launch(void* const* d_in, const int* in_sizes, int n_in,
                              void* d_out, int out_size, void* d_ws, size_t ws_size,
                              hipStream_t stream) {
    const float* u_emb = (const float*)d_in[0];
    const float* p_emb = (const float*)d_in[1];
    // d_in[2] = p_views (unused by reference math)
    const float* t_arr = (const float*)d_in[3];
    const int*   uid   = (const int*)d_in[4];
    const int*   pid   = (const int*)d_in[5];
    float* out = (float*)d_out;

    char* ws = (char*)d_ws;
    float* dist   = (float*)(ws);                                   // N floats
    int*   winner = (int*)  (ws + (size_t)NN * 4);                  // U*P ints
    float* scal   = (float*)(ws + (size_t)NN * 4 + (size_t)UU * PP * 4);   // 4 floats
    float* Opart  = (float*)(ws + (size_t)NN * 4 + (size_t)UU * PP * 4 + 256);

    // split-K width (power of two) that fits the workspace; deterministic in ws_size
    size_t fixed = (size_t)NN * 4 + (size_t)UU * PP * 4 + 256;
    int split = 256;
    while (split > 16 && fixed + (size_t)split * (DD * DD * 4) > ws_size) split >>= 1;
    int cpb = NCHUNKS / split;

    init_kernel<<<(UU * PP) / 256, 256, 0, stream>>>(winner, scal);
    fused_dist_gemm<<<split, 256, 0, stream>>>(u_emb, p_emb, uid, pid,
                                               dist, winner, Opart, cpb);
    reduce_ortho<<<(DD * DD) / 256, 256, 0, stream>>>(Opart, scal, split);
    loss_kernel<<<UU, 256, 0, stream>>>(t_arr, dist, winner, scal);
    final_kernel<<<1, 32, 0, stream>>>(scal, out);
}